// GraphModule_59012850647685
// MI455X (gfx1250) — compile-verified
//
#include <hip/hip_runtime.h>

typedef __attribute__((ext_vector_type(2))) float v2f;
typedef __attribute__((ext_vector_type(8))) float v8f;

#define D 64  // D_IN == D_OUT == 64

// ---------- degree init: deg[i] = 1 (the added self loop) ----------
__global__ void k_init_deg(float* __restrict__ deg, int n) {
    int i = blockIdx.x * blockDim.x + threadIdx.x;
    if (i < n) deg[i] = 1.0f;
}

// ---------- degree accumulate over non-self edges ----------
__global__ void k_deg(const int* __restrict__ src, const int* __restrict__ dst,
                      float* __restrict__ deg, int e_cnt) {
    int e = blockIdx.x * blockDim.x + threadIdx.x;
    if (e < e_cnt) {
        int s = src[e], t = dst[e];
        if (s != t) atomicAdd(deg + t, 1.0f);
    }
}

// ---------- dis = deg^-0.5 (deg >= 1 always, no zero case) ----------
__global__ void k_dis(const float* __restrict__ deg, float* __restrict__ dis, int n) {
    int i = blockIdx.x * blockDim.x + threadIdx.x;
    if (i < n) dis[i] = __frsqrt_rn(deg[i]);
}

// ---------- h = x @ W^T via V_WMMA_F32_16X16X4_F32 ----------
// One wave computes a 16-row x 64-col tile of h.
// A fragment (16x4 f32): lane L -> M = L%16, K pair = (L/16)*2 .. +1
// B fragment (4x16 f32): B[k][nc] = W[nc][k] -> same row-pair load from W[nc]
// D (16x16 f32, 8 VGPRs): vgpr r, lane L -> M = r + (L/16)*8, N = L%16
__global__ __launch_bounds__(128) void k_gemm(const float* __restrict__ x,
                                              const float* __restrict__ w,
                                              float* __restrict__ h, int n) {
    const int lane = threadIdx.x & 31;
    const int wv   = threadIdx.x >> 5;
    const int tile = blockIdx.x * 4 + wv;     // 16-row tile index
    const int row0 = tile * 16;
    const int m    = lane & 15;
    const int koff = (lane >> 4) * 2;         // 0 or 2

    int arow = row0 + m;
    if (arow > n - 1) arow = n - 1;           // clamp (keep EXEC all-1s for WMMA)
    const float* __restrict__ xp = x + arow * D;
    const float* __restrict__ w0 = w + (0 * 16 + m) * D;
    const float* __restrict__ w1 = w + (1 * 16 + m) * D;
    const float* __restrict__ w2 = w + (2 * 16 + m) * D;
    const float* __restrict__ w3 = w + (3 * 16 + m) * D;

    v8f acc0 = {}, acc1 = {}, acc2 = {}, acc3 = {};
#pragma unroll
    for (int k = 0; k < D; k += 4) {
        v2f a  = *(const v2f*)(xp + k + koff);
        v2f b0 = *(const v2f*)(w0 + k + koff);
        v2f b1 = *(const v2f*)(w1 + k + koff);
        v2f b2 = *(const v2f*)(w2 + k + koff);
        v2f b3 = *(const v2f*)(w3 + k + koff);
        acc0 = __builtin_amdgcn_wmma_f32_16x16x4_f32(false, a, false, b0, (short)0, acc0, false, false);
        acc1 = __builtin_amdgcn_wmma_f32_16x16x4_f32(false, a, false, b1, (short)0, acc1, false, false);
        acc2 = __builtin_amdgcn_wmma_f32_16x16x4_f32(false, a, false, b2, (short)0, acc2, false, false);
        acc3 = __builtin_amdgcn_wmma_f32_16x16x4_f32(false, a, false, b3, (short)0, acc3, false, false);
    }

    const int nc = lane & 15;
    const int mb = (lane >> 4) * 8;
#pragma unroll
    for (int r = 0; r < 8; ++r) {
        int row = row0 + mb + r;
        if (row < n) {
            float* __restrict__ hp = h + row * D + nc;
            hp[0]  = acc0[r];
            hp[16] = acc1[r];
            hp[32] = acc2[r];
            hp[48] = acc3[r];
        }
    }
}

// ---------- out[i][d] = h[i][d] * dis[i]^2 + bias[d]  (self-loop term + bias) ----------
__global__ void k_outinit(const float* __restrict__ h, const float* __restrict__ dis,
                          const float* __restrict__ bias, float* __restrict__ out, int n) {
    int gid = blockIdx.x * blockDim.x + threadIdx.x;
    if (gid < n * D) {
        int i = gid >> 6;
        int d = gid & (D - 1);
        float s = dis[i];
        out[gid] = h[gid] * s * s + bias[d];
    }
}

// ---------- edge scatter: out[dst] += dis[src]*dis[dst] * h[src] ----------
// 64 consecutive threads (2 waves) per edge: index loads coalesce, feature
// slices are 128B-contiguous per wave; f32 atomics resolve in the 192MB L2.
__global__ void k_scatter(const int* __restrict__ src, const int* __restrict__ dst,
                          const float* __restrict__ dis, const float* __restrict__ h,
                          float* __restrict__ out, int e_cnt) {
    long long gid = (long long)blockIdx.x * blockDim.x + threadIdx.x;
    int e = (int)(gid >> 6);
    if (e >= e_cnt) return;
    int d = (int)(gid & (D - 1));
    int s = src[e], t = dst[e];
    if (s == t) return;                         // filtered self loops: weight 0
    float coef = dis[s] * dis[t];
    atomicAdd(out + t * D + d, coef * h[s * D + d]);
}

extern "C" void kernel_launch(void* const* d_in, const int* in_sizes, int n_in,
                              void* d_out, int out_size, void* d_ws, size_t ws_size,
                              hipStream_t stream) {
    const float* x    = (const float*)d_in[0];   // [N, 64]
    const int*   ei   = (const int*)d_in[1];     // [2, E]
    const float* w    = (const float*)d_in[2];   // [64, 64]  W[n][k]
    const float* bias = (const float*)d_in[3];   // [64]
    float* out = (float*)d_out;

    const int n = in_sizes[0] / D;
    const int e = in_sizes[1] / 2;
    const int* src = ei;
    const int* dst = ei + e;

    // workspace layout: deg[n] | dis[n] | h[n*D]   (~26.4 MB)
    float* deg = (float*)d_ws;
    float* dis = deg + n;
    float* h   = dis + n;

    k_init_deg<<<(n + 255) / 256, 256, 0, stream>>>(deg, n);
    k_deg<<<(e + 255) / 256, 256, 0, stream>>>(src, dst, deg, e);

    const int ntiles = (n + 15) / 16;
    k_gemm<<<(ntiles + 3) / 4, 128, 0, stream>>>(x, w, h, n);

    k_dis<<<(n + 255) / 256, 256, 0, stream>>>(deg, dis, n);

    const long long tot = (long long)n * D;
    k_outinit<<<(int)((tot + 255) / 256), 256, 0, stream>>>(h, dis, bias, out, n);

    const long long etot = (long long)e * D;
    k_scatter<<<(int)((etot + 255) / 256), 256, 0, stream>>>(src, dst, dis, h, out, e);
}